// contrastAcrossSegments_75342316306682
// MI455X (gfx1250) — compile-verified
//
#include <hip/hip_runtime.h>

typedef __attribute__((ext_vector_type(2))) float v2f;
typedef __attribute__((ext_vector_type(8))) float v8f;

// Problem constants from the reference (B,N,M,D) = (8,64,32,256)
constexpr int Bsz  = 8;
constexpr int Nn   = 64;
constexpr int Mm   = 32;
constexpr int Dd   = 256;
constexpr int NM   = Nn * Mm;        // 2048
constexpr int NPOS = Bsz * NM;       // 16384 positions

// Workspace layout (floats): [0, NPOS)   denominator accumulators
//                            [NPOS+0]    sum(log term)
//                            [NPOS+1]    sum(1 - num)
//                            [NPOS+2]    count of non-padded positions

// ---------------------------------------------------------------------------
// Kernel A: gather-dot via WMMA f32 16x16x4.
// One wave per 16-sample tile. A rows = 16 gathered self vectors,
// B cols = 16 gathered cross vectors, K accumulated over D=256 in steps of 4.
// Only the diagonal of the 16x16 accumulator holds the wanted pairwise dots.
// sim = valid ? exp(dot) : 0, wave-reduced (all 16 samples of a tile share
// one denominator slot: tile>>3), then one global_atomic_add_f32.
// ---------------------------------------------------------------------------
__global__ __launch_bounds__(256) void denom_wmma_kernel(
    const float* __restrict__ self_u,             // (B*NM, D)
    const float* __restrict__ cross_u,            // (B*NM, D)
    const unsigned char* __restrict__ valid_mask, // (B, NM, NM) bool
    const int* __restrict__ inds,                 // (K, 3) [b, pos, neg]
    float* __restrict__ denom,                    // (NPOS) accumulators
    long nTiles)
{
  __shared__ float tileC[8][16 * 16];

  const int lane = threadIdx.x & 31;
  const int wv   = threadIdx.x >> 5;
  const long tile = (long)blockIdx.x * 8 + wv;
  if (tile >= nTiles) return;

  const int m  = lane & 15;   // sample slot within tile (= A row m / B col n)
  const int hi = lane >> 4;   // K sub-offset selector per f32 WMMA layout
  const long k = tile * 16 + m;

  const int b   = inds[3 * k + 0];
  const int pos = inds[3 * k + 1];
  const int neg = inds[3 * k + 2];

  // Prefetch the validity byte now so its latency hides under the WMMA loop.
  const unsigned char ok = valid_mask[((size_t)b * NM + pos) * NM + neg];

  // Lane L supplies A[m, 4k+2hi .. +1] and B[4k+2hi .. +1, m] each step.
  const float* pa = self_u  + ((size_t)b * NM + pos) * Dd + 2 * hi;
  const float* pc = cross_u + ((size_t)b * NM + neg) * Dd + 2 * hi;

  v8f acc = {};
#pragma unroll 4
  for (int kk = 0; kk < Dd / 4; ++kk) {
    v2f af = *(const v2f*)(pa + kk * 4);   // 8B-aligned global_load_b64
    v2f bf = *(const v2f*)(pc + kk * 4);
    acc = __builtin_amdgcn_wmma_f32_16x16x4_f32(
        /*neg_a=*/false, af, /*neg_b=*/false, bf,
        /*c_mod=*/(short)0, acc, /*reuse_a=*/false, /*reuse_b=*/false);
  }

  // Spill the 16x16 f32 accumulator to LDS; C layout: row = r + 8*hi, col = m.
#pragma unroll
  for (int r = 0; r < 8; ++r)
    tileC[wv][(r + 8 * hi) * 16 + m] = acc[r];
  asm volatile("s_wait_dscnt 0" ::: "memory");

  float sim = 0.f;
  if (hi == 0) {  // lanes 0..15 each pick one diagonal element
    float d = tileC[wv][m * 16 + m];
    sim = ok ? __expf(d) : 0.f;
  }
  // wave32 reduction (upper-half lanes contribute 0)
  for (int off = 16; off; off >>= 1) sim += __shfl_xor(sim, off, 32);

  if (lane == 0) atomicAdd(&denom[tile >> 3], sim);
}

// ---------------------------------------------------------------------------
// Kernel B1: one position per thread (64 blocks x 256 = 16384). Compute the
// positive dot, combine with the denominator, block-reduce the three partial
// sums, and atomically accumulate into the workspace accumulators.
// ---------------------------------------------------------------------------
__global__ __launch_bounds__(256) void partial_kernel(
    const float* __restrict__ self_u,
    const float* __restrict__ cross_u,
    const unsigned char* __restrict__ pad,   // (NPOS) bool
    const float* __restrict__ denom,
    float* __restrict__ accum)               // 3 floats at denom+NPOS
{
  __shared__ float sLog[256], sSim[256], sCnt[256];

  const int p = blockIdx.x * 256 + threadIdx.x;

  float sumLog = 0.f, sumSim = 0.f, cnt = 0.f;
  if (p < NPOS) {
    const float4* a4 = (const float4*)(self_u  + (size_t)p * Dd);
    const float4* c4 = (const float4*)(cross_u + (size_t)p * Dd);
    float4 acc = {0.f, 0.f, 0.f, 0.f};
#pragma unroll 8
    for (int i = 0; i < Dd / 4; ++i) {
      float4 a = a4[i], c = c4[i];
      acc.x += a.x * c.x; acc.y += a.y * c.y;
      acc.z += a.z * c.z; acc.w += a.w * c.w;
    }
    float num = acc.x + acc.y + acc.z + acc.w;
    if (!pad[p]) {
      float numer = __expf(num);
      float den   = denom[p];
      sumLog = __logf(numer / (numer + den));
      sumSim = 1.0f - num;
      cnt    = 1.0f;
    }
  }

  sLog[threadIdx.x] = sumLog;
  sSim[threadIdx.x] = sumSim;
  sCnt[threadIdx.x] = cnt;
  __syncthreads();
  for (int s = 128; s; s >>= 1) {
    if ((int)threadIdx.x < s) {
      sLog[threadIdx.x] += sLog[threadIdx.x + s];
      sSim[threadIdx.x] += sSim[threadIdx.x + s];
      sCnt[threadIdx.x] += sCnt[threadIdx.x + s];
    }
    __syncthreads();
  }
  if (threadIdx.x == 0) {
    atomicAdd(&accum[0], sLog[0]);
    atomicAdd(&accum[1], sSim[0]);
    atomicAdd(&accum[2], sCnt[0]);
  }
}

// ---------------------------------------------------------------------------
// Kernel B2: trivial epilogue.
// ---------------------------------------------------------------------------
__global__ void epilogue_kernel(const float* __restrict__ accum,
                                float* __restrict__ out)
{
  float c = accum[2];
  out[0] = -accum[0] / c;   // -log_exp_loss
  out[1] =  accum[1] / c;   // sim_loss
}

// ---------------------------------------------------------------------------
extern "C" void kernel_launch(void* const* d_in, const int* in_sizes, int n_in,
                              void* d_out, int out_size, void* d_ws, size_t ws_size,
                              hipStream_t stream) {
  const float* self_u  = (const float*)d_in[0];
  const float* cross_u = (const float*)d_in[1];
  const unsigned char* pad   = (const unsigned char*)d_in[2];  // bool mask
  const unsigned char* valid = (const unsigned char*)d_in[3];  // bool mask
  const int* inds = (const int*)d_in[4];                       // (K,3) int
  float* out   = (float*)d_out;
  float* denom = (float*)d_ws;                                 // NPOS floats
  float* accum = denom + NPOS;                                 // 3 floats

  long nK     = (long)in_sizes[4] / 3;   // number of sampled pairs
  long nTiles = nK / 16;                 // 16 samples per WMMA tile

  hipMemsetAsync(denom, 0, (NPOS + 3) * sizeof(float), stream);

  int blocks = (int)((nTiles + 7) / 8);  // 8 waves (tiles) per 256-thread block
  denom_wmma_kernel<<<blocks, 256, 0, stream>>>(self_u, cross_u, valid, inds,
                                                denom, nTiles);
  partial_kernel<<<NPOS / 256, 256, 0, stream>>>(self_u, cross_u, pad, denom,
                                                 accum);
  epilogue_kernel<<<1, 1, 0, stream>>>(accum, out);
}